// SceneModel_84189948936587
// MI455X (gfx1250) — compile-verified
//
#include <hip/hip_runtime.h>
#include <math.h>

typedef __attribute__((ext_vector_type(2))) float v2f;
typedef __attribute__((ext_vector_type(8))) float v8f;

#define N_POINTS  2048
#define WIDTH     64
#define HEIGHT    64
#define N_VIEWS   4
#define GAMMA     1.0f
#define ZMAX      45.0f
#define SHARPNESS 50.0f
#define BG_EPS    0.01f

// ---------------------------------------------------------------------------
// Prep: per (view, point) camera-space position + packed attributes.
// c_i = sum_j R[j][i] * (p - pos)_j  (i.e. R^T (p-pos)), then z negated.
// ---------------------------------------------------------------------------
__global__ void prep_kernel(const float* __restrict__ vert_pos,
                            const float* __restrict__ vert_col,
                            const float* __restrict__ vert_rad,
                            const float* __restrict__ cam,
                            float4* __restrict__ c4,
                            float4* __restrict__ col4) {
    int idx = blockIdx.x * blockDim.x + threadIdx.x;
    if (idx >= N_VIEWS * N_POINTS) return;
    int v = idx / N_POINTS;
    int n = idx - v * N_POINTS;

    const float* cv = cam + v * 8;
    float px = cv[0], py = cv[1], pz = cv[2];
    float rx = cv[3], ry = cv[4], rz = cv[5];

    float theta = sqrtf(rx * rx + ry * ry + rz * rz);
    float inv   = 1.0f / fmaxf(theta, 1e-8f);
    float kx = rx * inv, ky = ry * inv, kz = rz * inv;
    float ct = cosf(theta), st = sinf(theta), omc = 1.0f - ct;

    // R = c*I + s*K + (1-c)*k k^T
    float R00 = ct + omc * kx * kx;
    float R01 = -st * kz + omc * kx * ky;
    float R02 =  st * ky + omc * kx * kz;
    float R10 =  st * kz + omc * ky * kx;
    float R11 = ct + omc * ky * ky;
    float R12 = -st * kx + omc * ky * kz;
    float R20 = -st * ky + omc * kz * kx;
    float R21 =  st * kx + omc * kz * ky;
    float R22 = ct + omc * kz * kz;

    float qx = vert_pos[n * 3 + 0] - px;
    float qy = vert_pos[n * 3 + 1] - py;
    float qz = vert_pos[n * 3 + 2] - pz;

    float cx = R00 * qx + R10 * qy + R20 * qz;
    float cy = R01 * qx + R11 * qy + R21 * qz;
    float cz = -(R02 * qx + R12 * qy + R22 * qz);   // * [1,1,-1]
    float c2 = cx * cx + cy * cy + cz * cz;

    c4[v * N_POINTS + n] = make_float4(cx, cy, cz, c2);
    if (v == 0) {
        col4[n] = make_float4(vert_col[n * 3 + 0], vert_col[n * 3 + 1],
                              vert_col[n * 3 + 2], vert_rad[n]);
    }
}

// ---------------------------------------------------------------------------
// Render: one wave32 per (view, 16-pixel tile).
// WMMA f32 16x16x4:  t[point(M) x pixel(N)] = c(16x4) x d(4x16).
// D layout: lane holds pixel N = lane&15, rows r -> points tbase + r + 8*half.
// ---------------------------------------------------------------------------
__global__ void render_kernel(const float* __restrict__ cam,
                              const float4* __restrict__ c4,
                              const float4* __restrict__ col4,
                              float* __restrict__ out) {
    const int lane   = threadIdx.x & 31;
    const int waveId = blockIdx.x * (blockDim.x >> 5) + (threadIdx.x >> 5);
    const int v      = waveId >> 8;     // 256 tiles per view
    const int tile   = waveId & 255;
    const int half   = lane >> 4;       // 0: lanes 0-15, 1: lanes 16-31
    const int m      = lane & 15;       // pixel column / A-point row

    const int pix = tile * 16 + m;      // flattened pixel 0..4095
    const int hh  = pix >> 6;
    const int ww  = pix & 63;

    const float* cv = cam + v * 8;
    const float focal = cv[6], sensor = cv[7];

    // Normalized ray direction for pixel m of this tile.
    float us = ((float)ww + 0.5f) / (float)WIDTH - 0.5f;
    float vs = (((float)hh + 0.5f) / (float)HEIGHT - 0.5f) *
               ((float)HEIGHT / (float)WIDTH);
    float dx = sensor * us;
    float dy = -sensor * vs;
    float dz = focal;
    float rn = __builtin_amdgcn_rsqf(dx * dx + dy * dy + dz * dz);
    dx *= rn; dy *= rn; dz *= rn;

    // B operand (4x16, K x pixel): lanes 0-15 -> K=0,1 ; lanes 16-31 -> K=2,3
    v2f B;
    B.x = half ? dz : dx;
    B.y = half ? 0.0f : dy;

    const float4* c4v = c4 + v * N_POINTS;

    const float inv_zmax = 1.0f / (GAMMA * ZMAX);   // compile-time constant

    float accR = 0.0f, accG = 0.0f, accB = 0.0f, accD = 0.0f;

    for (int tbase = 0; tbase < N_POINTS; tbase += 16) {
        // A operand (16x4, point x K): lane's point row = m.
        float4 ca = c4v[tbase + m];
        v2f A;
        A.x = half ? ca.z : ca.x;
        A.y = half ? 0.0f : ca.y;

        v8f c = {};
        v8f t = __builtin_amdgcn_wmma_f32_16x16x4_f32(
            /*neg_a=*/false, A, /*neg_b=*/false, B,
            /*c_mod=*/(short)0, c, /*reuse_a=*/false, /*reuse_b=*/false);

        const int pbase = tbase + half * 8;
#pragma unroll
        for (int r = 0; r < 8; ++r) {
            float4 cc = c4v[pbase + r];      // .w = |c|^2
            float4 cl = col4[pbase + r];     // rgb + rad

            float tv   = t[r];
            float diff = fmaxf(cc.w - tv * tv, 1e-8f);
            float dist = __builtin_amdgcn_sqrtf(diff);        // native v_sqrt_f32
            float x    = (cl.w - dist) * SHARPNESS;
            // sigmoid via native rcp (denominator >= 1, well conditioned)
            float hit  = __builtin_amdgcn_rcpf(1.0f + __expf(-x));
            float tc   = fminf(fmaxf(tv, 0.0f), ZMAX);
            float dw   = __expf((ZMAX - tc) * inv_zmax);      // mul, not div
            float w    = (tv > 0.0f) ? hit * dw : 0.0f;

            accR += w * cl.x;
            accG += w * cl.y;
            accB += w * cl.z;
            accD += w;
        }
    }

    // Combine the two half-wave partial sums (points split 8/8 per tile).
    accR += __shfl_xor(accR, 16, 32);
    accG += __shfl_xor(accG, 16, 32);
    accB += __shfl_xor(accB, 16, 32);
    accD += __shfl_xor(accD, 16, 32);

    if (half == 0) {
        float den = accD + __expf(BG_EPS / GAMMA);
        float inv = __builtin_amdgcn_rcpf(den);
        int o = (v * HEIGHT * WIDTH + pix) * 3;
        out[o + 0] = accR * inv;
        out[o + 1] = accG * inv;
        out[o + 2] = accB * inv;
    }
}

// ---------------------------------------------------------------------------
extern "C" void kernel_launch(void* const* d_in, const int* in_sizes, int n_in,
                              void* d_out, int out_size, void* d_ws, size_t ws_size,
                              hipStream_t stream) {
    const float* vert_pos = (const float*)d_in[0];   // (1, 2048, 3)
    const float* vert_col = (const float*)d_in[1];   // (1, 2048, 3)
    const float* vert_rad = (const float*)d_in[2];   // (1, 2048)
    const float* cam      = (const float*)d_in[3];   // (4, 8)
    float* out = (float*)d_out;                      // (4, 64, 64, 3)

    float4* c4   = reinterpret_cast<float4*>(d_ws);          // 4*2048 float4
    float4* col4 = c4 + N_VIEWS * N_POINTS;                  // 2048 float4

    // Prep: 4 views * 2048 points
    prep_kernel<<<(N_VIEWS * N_POINTS + 255) / 256, 256, 0, stream>>>(
        vert_pos, vert_col, vert_rad, cam, c4, col4);

    // Render: 4 views * 256 tiles = 1024 waves -> 128 blocks of 8 waves
    render_kernel<<<128, 256, 0, stream>>>(cam, c4, col4, out);
}